// Wasserstein_360777253563
// MI455X (gfx1250) — compile-verified
//
#include <hip/hip_runtime.h>

typedef float    v2f __attribute__((ext_vector_type(2)));
typedef float    v8f __attribute__((ext_vector_type(8)));
typedef _Float16 h8  __attribute__((ext_vector_type(8)));

#define SHIFT 16.0f        // K' = exp(SHIFT - C); exactly neutral for Sinkhorn
#define AB_CONST (1.0f/1024.0f)

// ---------------------------------------------------------------------------
// Shared WMMA tile GEMM: acc[cc][j] = sum_d X[row][d] * Y[col][d]
// Wave computes a 16x64 strip of cross = X @ Y^T for one l, f32 WMMA 16x16x4.
// A layout: lane(0-15)=M, vgpr{0,1}=K{0,1}; lane(16-31): K{2,3}. B mirrors with N.
// ---------------------------------------------------------------------------
__device__ __forceinline__ void cost_tile_gemm(
    const float* __restrict__ X, const float* __restrict__ Y,
    int l, int r0, int c0, int lid, int half, v8f acc[4]) {
  const float* xrow = X + (size_t)(r0 + lid) * 8192 + l * 128 + 2 * half;
  const float* yb   = Y + (size_t)(c0 + lid) * 8192 + l * 128 + 2 * half;
  const float* yb0 = yb;
  const float* yb1 = yb + (size_t)16 * 8192;
  const float* yb2 = yb + (size_t)32 * 8192;
  const float* yb3 = yb + (size_t)48 * 8192;
#pragma unroll 4
  for (int d0 = 0; d0 < 128; d0 += 4) {
    v2f a  = *reinterpret_cast<const v2f*>(xrow + d0);
    v2f b0 = *reinterpret_cast<const v2f*>(yb0 + d0);
    v2f b1 = *reinterpret_cast<const v2f*>(yb1 + d0);
    v2f b2 = *reinterpret_cast<const v2f*>(yb2 + d0);
    v2f b3 = *reinterpret_cast<const v2f*>(yb3 + d0);
    acc[0] = __builtin_amdgcn_wmma_f32_16x16x4_f32(false, a, false, b0, (short)0, acc[0], false, false);
    acc[1] = __builtin_amdgcn_wmma_f32_16x16x4_f32(false, a, false, b1, (short)0, acc[1], false, false);
    acc[2] = __builtin_amdgcn_wmma_f32_16x16x4_f32(false, a, false, b2, (short)0, acc[2], false, false);
    acc[3] = __builtin_amdgcn_wmma_f32_16x16x4_f32(false, a, false, b3, (short)0, acc[3], false, false);
  }
}

// ---------------------------------------------------------------------------
// Row norms: X2[l][n] = sum_d X[n][l][d]^2 (and same for Y). One row per wave.
// ---------------------------------------------------------------------------
__global__ void norms_kernel(const float* __restrict__ X, const float* __restrict__ Y,
                             float* __restrict__ X2, float* __restrict__ Y2) {
  int wave = threadIdx.x >> 5, lane = threadIdx.x & 31;
  int rowId = blockIdx.x * 8 + wave;           // 0 .. 131071
  const float* src = X; float* dst = X2;
  int idx = rowId;
  if (idx >= 65536) { src = Y; dst = Y2; idx -= 65536; }
  int l = idx >> 10;
  int n = idx & 1023;
  float4 v = *reinterpret_cast<const float4*>(src + (size_t)n * 8192 + l * 128 + lane * 4);
  float s = v.x * v.x + v.y * v.y + v.z * v.z + v.w * v.w;
  for (int m = 16; m >= 1; m >>= 1) s += __shfl_xor(s, m, 32);
  if (lane == 0) dst[l * 1024 + n] = s;
}

__global__ void init_v_kernel(float* __restrict__ v) {
  int i = blockIdx.x * 256 + threadIdx.x;
  v[i] = 1.0f;
}

// ---------------------------------------------------------------------------
// Phase 1: K'[l][n][m] = fp16(exp(SHIFT - sqrt(x2+y2-2*cross))).
// Block = 8 waves; wave -> 16x64 strip. Grid = 64 * 8 * 16 = 8192.
// ---------------------------------------------------------------------------
__global__ void computeK_kernel(const float* __restrict__ X, const float* __restrict__ Y,
                                const float* __restrict__ X2, const float* __restrict__ Y2,
                                _Float16* __restrict__ Kp) {
  int wave = threadIdx.x >> 5, lane = threadIdx.x & 31;
  int bx = blockIdx.x;
  int l   = bx >> 7;
  int rem = bx & 127;
  int r0  = (rem >> 4) * 128 + wave * 16;
  int c0  = (rem & 15) * 64;
  int half = lane >> 4, lid = lane & 15;

  v8f acc[4] = {v8f{}, v8f{}, v8f{}, v8f{}};
  cost_tile_gemm(X, Y, l, r0, c0, lid, half, acc);

  int rowoff = r0 + 8 * half;
#pragma unroll
  for (int cc = 0; cc < 4; ++cc) {
    int col = c0 + cc * 16 + lid;
    float y2v = Y2[l * 1024 + col];
#pragma unroll
    for (int j = 0; j < 8; ++j) {
      int row = rowoff + j;
      float c2 = X2[l * 1024 + row] + y2v - 2.0f * acc[cc][j];
      float C  = sqrtf(fmaxf(c2, 0.0f));
      Kp[((size_t)(l * 1024 + row) << 10) + col] = (_Float16)__expf(SHIFT - C);
    }
  }
}

// ---------------------------------------------------------------------------
// u[l][n] = a / sum_m K'[l][n][m] * v[l][m].   One row per wave. Grid = 8192.
// ---------------------------------------------------------------------------
__global__ void u_update_kernel(const _Float16* __restrict__ Kp,
                                const float* __restrict__ v, float* __restrict__ u) {
  int wave = threadIdx.x >> 5, lane = threadIdx.x & 31;
  int bx = blockIdx.x;
  int l = bx >> 7;
  int n = (bx & 127) * 8 + wave;
  const _Float16* krow = Kp + ((size_t)(l * 1024 + n) << 10);
  const float* vrow = v + l * 1024;
  float s = 0.0f;
#pragma unroll
  for (int k = 0; k < 4; ++k) {
    int m0 = k * 256 + lane * 8;
    h8 kv = *reinterpret_cast<const h8*>(krow + m0);
    float4 v0 = *reinterpret_cast<const float4*>(vrow + m0);
    float4 v1 = *reinterpret_cast<const float4*>(vrow + m0 + 4);
    s += (float)kv[0] * v0.x + (float)kv[1] * v0.y + (float)kv[2] * v0.z + (float)kv[3] * v0.w
       + (float)kv[4] * v1.x + (float)kv[5] * v1.y + (float)kv[6] * v1.z + (float)kv[7] * v1.w;
  }
  for (int m = 16; m >= 1; m >>= 1) s += __shfl_xor(s, m, 32);
  if (lane == 0) u[l * 1024 + n] = AB_CONST / s;
}

// ---------------------------------------------------------------------------
// v[l][m] = b / sum_n u[l][n] * K'[l][n][m].  Block: 64 cols x 4 row-chunks.
// Grid = 64 * 16 = 1024.
// ---------------------------------------------------------------------------
__global__ void v_update_kernel(const _Float16* __restrict__ Kp,
                                const float* __restrict__ u, float* __restrict__ v) {
  __shared__ float part[4][64];
  int tx = threadIdx.x & 63;
  int ty = threadIdx.x >> 6;
  int bx = blockIdx.x;
  int l = bx >> 4;
  int m = (bx & 15) * 64 + tx;
  const _Float16* kcol = Kp + ((size_t)l << 20) + m;
  const float* urow = u + l * 1024;
  float s = 0.0f;
  int n0 = ty * 256;
#pragma unroll 4
  for (int n = n0; n < n0 + 256; ++n)
    s += urow[n] * (float)kcol[(size_t)n << 10];
  part[ty][tx] = s;
  __syncthreads();
  if (ty == 0) {
    float t = part[0][tx] + part[1][tx] + part[2][tx] + part[3][tx];
    v[l * 1024 + m] = AB_CONST / t;
  }
}

// ---------------------------------------------------------------------------
// Final: recompute C via WMMA, P = u*K'*v, accumulate sum(P*C) per block.
// Grid = 8192 (same mapping as computeK).
// ---------------------------------------------------------------------------
__global__ void final_kernel(const float* __restrict__ X, const float* __restrict__ Y,
                             const float* __restrict__ X2, const float* __restrict__ Y2,
                             const float* __restrict__ u, const float* __restrict__ v,
                             float* __restrict__ partials) {
  __shared__ float wsum[8];
  int wave = threadIdx.x >> 5, lane = threadIdx.x & 31;
  int bx = blockIdx.x;
  int l   = bx >> 7;
  int rem = bx & 127;
  int r0  = (rem >> 4) * 128 + wave * 16;
  int c0  = (rem & 15) * 64;
  int half = lane >> 4, lid = lane & 15;

  v8f acc[4] = {v8f{}, v8f{}, v8f{}, v8f{}};
  cost_tile_gemm(X, Y, l, r0, c0, lid, half, acc);

  int rowoff = r0 + 8 * half;
  float uv[8], x2v[8];
#pragma unroll
  for (int j = 0; j < 8; ++j) {
    uv[j]  = u[l * 1024 + rowoff + j];
    x2v[j] = X2[l * 1024 + rowoff + j];
  }
  float psum = 0.0f;
#pragma unroll
  for (int cc = 0; cc < 4; ++cc) {
    int col = c0 + cc * 16 + lid;
    float y2v = Y2[l * 1024 + col];
    float vv  = v[l * 1024 + col];
#pragma unroll
    for (int j = 0; j < 8; ++j) {
      float c2 = x2v[j] + y2v - 2.0f * acc[cc][j];
      float C  = sqrtf(fmaxf(c2, 0.0f));
      float p  = uv[j] * __expf(SHIFT - C) * vv;
      psum += p * C;
    }
  }
  for (int m = 16; m >= 1; m >>= 1) psum += __shfl_xor(psum, m, 32);
  if (lane == 0) wsum[wave] = psum;
  __syncthreads();
  if (threadIdx.x == 0) {
    float t = 0.0f;
#pragma unroll
    for (int i = 0; i < 8; ++i) t += wsum[i];
    partials[bx] = t;
  }
}

__global__ void reduce_kernel(const float* __restrict__ partials, float* __restrict__ out) {
  __shared__ float sh[256];
  float s = 0.0f;
  for (int i = threadIdx.x; i < 8192; i += 256) s += partials[i];
  sh[threadIdx.x] = s;
  __syncthreads();
  for (int st = 128; st >= 1; st >>= 1) {
    if (threadIdx.x < st) sh[threadIdx.x] += sh[threadIdx.x + st];
    __syncthreads();
  }
  if (threadIdx.x == 0) out[0] = sh[0] * (1.0f / (1024.0f * 1024.0f));
}

// ---------------------------------------------------------------------------
extern "C" void kernel_launch(void* const* d_in, const int* in_sizes, int n_in,
                              void* d_out, int out_size, void* d_ws, size_t ws_size,
                              hipStream_t stream) {
  (void)in_sizes; (void)n_in; (void)out_size; (void)ws_size;
  const float* X = (const float*)d_in[0];   // (1024, 64, 128) f32
  const float* Y = (const float*)d_in[1];   // (1024, 64, 128) f32

  char* ws = (char*)d_ws;
  _Float16* Kp = (_Float16*)ws;                       // 64*1024*1024 fp16 = 128 MB
  float* X2 = (float*)(ws + (size_t)134217728);       // 64*1024
  float* Y2 = X2 + 65536;
  float* u  = Y2 + 65536;
  float* v  = u  + 65536;
  float* partials = v + 65536;                        // 8192 floats

  norms_kernel   <<<16384, 256, 0, stream>>>(X, Y, X2, Y2);
  init_v_kernel  <<<256,   256, 0, stream>>>(v);
  computeK_kernel<<<8192,  256, 0, stream>>>(X, Y, X2, Y2, Kp);

  for (int it = 0; it < 100; ++it) {
    u_update_kernel<<<8192, 256, 0, stream>>>(Kp, v, u);
    v_update_kernel<<<1024, 256, 0, stream>>>(Kp, u, v);
  }
  u_update_kernel<<<8192, 256, 0, stream>>>(Kp, v, u);

  final_kernel <<<8192, 256, 0, stream>>>(X, Y, X2, Y2, u, v, partials);
  reduce_kernel<<<1,    256, 0, stream>>>(partials, (float*)d_out);
}